// CropBboxesLayer_79413945303923
// MI455X (gfx1250) — compile-verified
//
#include <hip/hip_runtime.h>
#include <hip/hip_bf16.h>

// Crop-and-resize (bilinear, half-pixel centers), matching the JAX reference.
//
// Two-phase plan (memory-bound op, ~0.3 flop/byte -> no WMMA; minimize VALU
// so it stays under the 23.3 TB/s HBM roofline):
//   Phase 1 (63,488 threads): per-(crop,row) table {rowbase_lo, rowbase_hi,
//     wy, valid} and per-crop params {sw, rx=wf/sw, wf-1, w-1}. All four f32
//     divisions of the reference happen HERE (once per crop/row), not in the
//     12.7M-thread hot kernel.
//   Phase 2 (hot): one thread per output pixel; broadcast table reads, one
//     fma of column math, 4x b96 tap gathers, 12-FMA blend, b96 coalesced
//     store, gfx1250 global_prefetch_b8 for the next source row.

namespace {

constexpr int B  = 32;
constexpr int H  = 768;
constexpr int W  = 768;
constexpr int C  = 3;
constexpr int N  = 64;
constexpr int TH = 31;
constexpr int TW = 200;

constexpr int NCROP        = B * N;        // 2048
constexpr int PIX_PER_CROP = TH * TW;      // 6200
constexpr int ROW_ENTRIES  = NCROP * TH;   // 63488

// d_ws layout (16B-aligned int4 entries):
//   [0, ROW_ENTRIES)                : row table, int4 per (crop, i)
//   [ROW_ENTRIES, ROW_ENTRIES+NCROP): crop params, int4 per crop
// total = (63488 + 2048) * 16 B = 1 MiB.

// ---------------------------------------------------------------- phase 1 --
__global__ __launch_bounds__(256) void precompute_tables_kernel(
    const int* __restrict__ bboxes,   // [B,N,4] i32 (y,x,h,w) pre-doubling
    int4*      __restrict__ rowtab,   // [NCROP*TH]
    int4*      __restrict__ croptab)  // [NCROP]
{
  const int t = blockIdx.x * blockDim.x + threadIdx.x;
  if (t >= ROW_ENTRIES) return;

  const int i    = t % TH;
  const int crop = t / TH;

  const int* bb = bboxes + 4 * crop;
  const int y0 = bb[0] * 2;
  const int x0 = bb[1] * 2;
  const int h  = bb[2] * 2;
  const int w  = bb[3] * 2;

  const float hf = (float)h;
  const float wf = (float)w;

  const float scale = fminf((float)TW / wf, (float)TH / hf);
  int sh = (int)floorf(hf * scale); if (sh < 1) sh = 1;
  int sw = (int)floorf(wf * scale); if (sw < 1) sw = 1;

  if (i == 0) {  // one writer per crop
    int4 cp;
    cp.x = sw;
    cp.y = __float_as_int(wf / (float)sw);  // rx
    cp.z = __float_as_int(wf - 1.0f);
    cp.w = w - 1;
    croptab[crop] = cp;
  }

  int4 re;
  const int ri = i - (TH - sh);
  if (ri < 0) {
    re.x = 0; re.y = 0; re.z = 0; re.w = 0;        // padded (invalid) row
  } else {
    const float ry = hf / (float)sh;
    float sy = ((float)ri + 0.5f) * ry - 0.5f;
    sy = fminf(fmaxf(sy, 0.0f), hf - 1.0f);
    const int   ylo = (int)floorf(sy);
    const int   yhi = min(ylo + 1, h - 1);
    const float wy  = sy - (float)ylo;
    re.x = ((y0 + ylo) * W + x0) * C;              // rowbase_lo (floats)
    re.y = ((y0 + yhi) * W + x0) * C;              // rowbase_hi (floats)
    re.z = __float_as_int(wy);
    re.w = 1;
  }
  rowtab[t] = re;
}

// ---------------------------------------------------------------- phase 2 --
__global__ __launch_bounds__(256) void crop_resize_gather_kernel(
    const float* __restrict__ images,   // [B,H,W,C] f32
    const int4*  __restrict__ rowtab,
    const int4*  __restrict__ croptab,
    float*       __restrict__ out)      // [B,N,TH,TW,C] f32
{
  const int tid  = blockIdx.x * blockDim.x + threadIdx.x;   // pixel in crop
  const int crop = blockIdx.y;
  if (tid >= PIX_PER_CROP) return;

  const int i = tid / TW;         // 32-bit magic-mul, cheap
  const int j = tid - i * TW;

  const int4 re = rowtab[crop * TH + i];   // broadcast within a row
  const int4 cp = croptab[crop];           // broadcast within the crop

  float* o = out + ((long long)crop * PIX_PER_CROP + tid) * C;

  if ((re.w == 0) | (j >= cp.x)) {         // top padding or right padding
    o[0] = 0.0f; o[1] = 0.0f; o[2] = 0.0f;
    return;
  }

  // Column math (the only per-thread interpolation setup; no divisions).
  const float rx   = __int_as_float(cp.y);
  const float wfm1 = __int_as_float(cp.z);
  float sx = ((float)j + 0.5f) * rx - 0.5f;
  sx = fminf(fmaxf(sx, 0.0f), wfm1);
  const int   xlo = (int)floorf(sx);
  const int   xhi = min(xlo + 1, cp.w);
  const float wx  = sx - (float)xlo;

  const float* img = images + (long long)(crop / N) * (H * W * C);
  const float* p00 = img + re.x + xlo * C;
  const float* p01 = img + re.x + xhi * C;
  const float* p10 = img + re.y + xlo * C;
  const float* p11 = img + re.y + xhi * C;

  // gfx1250 prefetch (no LOADcnt): pre-touch next source row for this crop.
  __builtin_prefetch(p10 + (long long)W * C, 0, 1);

  const float wy   = __int_as_float(re.z);
  const float omwx = 1.0f - wx;
  const float omwy = 1.0f - wy;

#pragma unroll
  for (int c = 0; c < C; ++c) {
    const float top = p00[c] * omwx + p01[c] * wx;
    const float bot = p10[c] * omwx + p11[c] * wx;
    o[c] = top * omwy + bot * wy;
  }
}

} // anonymous namespace

extern "C" void kernel_launch(void* const* d_in, const int* in_sizes, int n_in,
                              void* d_out, int out_size, void* d_ws, size_t ws_size,
                              hipStream_t stream) {
  (void)in_sizes; (void)n_in; (void)out_size; (void)ws_size;

  const float* images = (const float*)d_in[0];   // [B,H,W,C] f32
  const int*   bboxes = (const int*)  d_in[1];   // [B,N,4]   i32
  float*       out    = (float*)      d_out;     // [B,N,TH,TW,C] f32

  int4* rowtab  = (int4*)d_ws;                   // 63488 * 16 B
  int4* croptab = rowtab + ROW_ENTRIES;          //  2048 * 16 B  (1 MiB total)

  // Phase 1: per-crop / per-row parameter tables (all divisions live here).
  {
    const int threads = 256;
    const int blocks  = (ROW_ENTRIES + threads - 1) / threads;   // 248
    precompute_tables_kernel<<<blocks, threads, 0, stream>>>(bboxes, rowtab, croptab);
  }

  // Phase 2: gather/blend/store, one thread per output pixel.
  {
    const int  threads = 256;                      // 8 wave32 waves
    const dim3 grid((PIX_PER_CROP + threads - 1) / threads,  // 25
                    NCROP);                                  // 2048 crops
    crop_resize_gather_kernel<<<grid, threads, 0, stream>>>(images, rowtab, croptab, out);
  }
}